// AdaptiveEdgeSparsifier_60722247631695
// MI455X (gfx1250) — compile-verified
//
#include <hip/hip_runtime.h>
#include <stdint.h>

// Row-wise top-k keep/zero (k = 1433 of 2048) via 4-pass 8-bit radix select.
// CDNA5 path: rows staged global->LDS with GLOBAL_LOAD_ASYNC_TO_LDS_B128
// (ASYNCcnt / s_wait_asynccnt), wave32 shuffle suffix-scan for bin selection.
//
// Memory-bound: 268 MB total traffic @ 23.3 TB/s => ~11.5 us floor; compute
// (~25 int ops/elem) is far under the VALU roofline, so single-touch HBM
// access is the optimization target. No matmul FLOPs exist => WMMA N/A.

#define TPB    256
#define N_COLS 2048
#define VPT    8           // values per thread (2048 / 256)

__device__ __forceinline__ uint32_t f2key(float f) {
  // Order-preserving float -> uint mapping (larger float => larger uint)
  uint32_t u = __float_as_uint(f);
  return (u & 0x80000000u) ? ~u : (u | 0x80000000u);
}

__global__ __launch_bounds__(TPB)
void adaptive_edge_sparsify_kernel(const float* __restrict__ in,
                                   float* __restrict__ out, int K) {
  __shared__ float    row[N_COLS];      // 8 KB staged row
  __shared__ uint32_t hist[TPB];        // 256-bin histogram
  __shared__ uint32_t waveSums[8];      // per-wave suffix totals
  __shared__ uint32_t sel_prefix;       // accumulated threshold key bits
  __shared__ uint32_t sel_remaining;    // rank remaining (from the top)

  const int tid  = threadIdx.x;
  const int lane = tid & 31;            // wave32 lane
  const int w    = tid >> 5;            // wave index (8 waves)

  const long long rowIdx = blockIdx.x;
  const float* gRow = in  + rowIdx * (long long)N_COLS;
  float*       gOut = out + rowIdx * (long long)N_COLS;

  // ---- Stage row into LDS: async global->LDS 16B copies (CDNA5 TDM-lite) ---
  {
    uint32_t ldsBase = (uint32_t)(uintptr_t)(&row[0]);   // low 32b = LDS offset
    uint32_t l0 = ldsBase + (uint32_t)tid * 16u;
    uint32_t l1 = l0 + 4096u;
    uint64_t g0 = (uint64_t)(uintptr_t)(gRow + tid * 4);
    uint64_t g1 = g0 + 4096u;
    asm volatile("global_load_async_to_lds_b128 %0, %1, off"
                 :: "v"(l0), "v"(g0) : "memory");
    asm volatile("global_load_async_to_lds_b128 %0, %1, off"
                 :: "v"(l1), "v"(g1) : "memory");
    asm volatile("s_wait_asynccnt 0x0" ::: "memory");
  }

  if (tid == 0) { sel_prefix = 0u; sel_remaining = (uint32_t)K; }
  __syncthreads();

  // ---- Pull this thread's 8 values into registers ----
  float4 v0 = reinterpret_cast<const float4*>(row)[tid];
  float4 v1 = reinterpret_cast<const float4*>(row)[tid + TPB];
  float    vals[VPT] = {v0.x, v0.y, v0.z, v0.w, v1.x, v1.y, v1.z, v1.w};
  uint32_t keys[VPT];
#pragma unroll
  for (int e = 0; e < VPT; ++e) keys[e] = f2key(vals[e]);
  uint32_t cand = 0xFFu;   // all 8 elements start as rank candidates

  // ---- 4-pass MSB-first radix select of the K-th largest key ----
#pragma unroll
  for (int pass = 0; pass < 4; ++pass) {
    const int shift = 24 - 8 * pass;

    hist[tid] = 0u;
    __syncthreads();

#pragma unroll
    for (int e = 0; e < VPT; ++e)
      if (cand & (1u << e))
        atomicAdd(&hist[(keys[e] >> shift) & 0xFFu], 1u);
    __syncthreads();

    // Suffix sum over bins >= tid.  Within-wave: Hillis-Steele via shfl_down
    uint32_t cnt = hist[tid];
    uint32_t v   = cnt;
#pragma unroll
    for (int d = 1; d < 32; d <<= 1) {
      uint32_t o = __shfl_down(v, d, 32);
      if (lane + d < 32) v += o;
    }
    if (lane == 0) waveSums[w] = v;       // lane0 holds this wave's 32-bin sum
    __syncthreads();

    uint32_t above = 0u;
    for (int w2 = w + 1; w2 < 8; ++w2) above += waveSums[w2];
    uint32_t S = v + above;               // #candidates with bin >= tid
    uint32_t r = sel_remaining;
    __syncthreads();

    // Unique thread whose bin contains the remaining rank (from the top)
    if (S >= r && (S - cnt) < r) {
      sel_prefix   |= ((uint32_t)tid) << shift;
      sel_remaining = r - (S - cnt);
    }
    __syncthreads();

    uint32_t selDigit = (sel_prefix >> shift) & 0xFFu;
#pragma unroll
    for (int e = 0; e < VPT; ++e)
      if (((keys[e] >> shift) & 0xFFu) != selDigit) cand &= ~(1u << e);
  }

  // ---- Apply mask: keep values whose key >= threshold key ----
  const uint32_t T = sel_prefix;          // key of the K-th largest element
  float4 o0, o1;
  o0.x = (keys[0] >= T) ? vals[0] : 0.0f;
  o0.y = (keys[1] >= T) ? vals[1] : 0.0f;
  o0.z = (keys[2] >= T) ? vals[2] : 0.0f;
  o0.w = (keys[3] >= T) ? vals[3] : 0.0f;
  o1.x = (keys[4] >= T) ? vals[4] : 0.0f;
  o1.y = (keys[5] >= T) ? vals[5] : 0.0f;
  o1.z = (keys[6] >= T) ? vals[6] : 0.0f;
  o1.w = (keys[7] >= T) ? vals[7] : 0.0f;
  reinterpret_cast<float4*>(gOut)[tid]       = o0;   // b128 coalesced stores
  reinterpret_cast<float4*>(gOut)[tid + TPB] = o1;
}

extern "C" void kernel_launch(void* const* d_in, const int* in_sizes, int n_in,
                              void* d_out, int out_size, void* d_ws, size_t ws_size,
                              hipStream_t stream) {
  const float* adj = (const float*)d_in[0];
  float*       out = (float*)d_out;

  const long long total = (long long)in_sizes[0];   // 8 * 2048 * 2048
  const int rows = (int)(total / N_COLS);           // 16384 rows
  int K = (int)((double)N_COLS * (1.0 - 0.3));      // int(2048*0.7) = 1433
  if (K < 1) K = 1;

  adaptive_edge_sparsify_kernel<<<rows, TPB, 0, stream>>>(adj, out, K);
}